// MeanAggregator_45913200394470
// MI455X (gfx1250) — compile-verified
//
#include <hip/hip_runtime.h>

// MeanAggregator on MI455X (gfx1250, wave32).
// out[b,:] = mean over 33 rows {neighbors[b,0..31], nodes[b]} of feat_table.
// Memory-bound gather (~338 MB reads; table is L2-resident at 51 MB vs 192 MB L2).
// Reduction is expressed exactly in f32 via V_WMMA_F32_16X16X4_F32:
//   D(16x16) += A(16 dims x 4 rows) * B(4 x 16 broadcast weights), w = 1/33.

typedef float v2f __attribute__((ext_vector_type(2)));
typedef float v8f __attribute__((ext_vector_type(8)));

#define D_FEAT 128
#define K_NBR  32

__global__ __launch_bounds__(256) void MeanAggregator_45913200394470_kernel(
    const int*   __restrict__ nodes,      // [B]
    const int*   __restrict__ neighbors,  // [B, 32]
    const float* __restrict__ feat,       // [N_NODES, 128]
    float*       __restrict__ out,        // [B, 128]
    int batch)
{
    const int lane = threadIdx.x & 31;
    const int wave = threadIdx.x >> 5;
    const int b    = blockIdx.x * (blockDim.x >> 5) + wave;
    if (b >= batch) return;               // wave-uniform guard: EXEC stays all-1

    const int m = lane & 15;              // dim-within-chunk (A's M row, per ISA layout)
    const int h = lane >> 4;              // half-wave selects K pair {2h, 2h+1}

    // one neighbor index per lane
    const int nbr  = neighbors[(long long)b * K_NBR + lane];
    const int self = nodes[b];

    const float w = 1.0f / (float)(K_NBR + 1);   // 1/33

    v8f acc[8] = {};                      // 8 x (16x16 f32 C/D) accumulators = 64 VGPRs

    // B operand for dense chunks: every weight = w
    v2f bAll;
    bAll[0] = w;
    bAll[1] = w;

    // chunks c = 0..7 cover neighbor rows 4c .. 4c+3 (exactly the 32 neighbors)
#pragma unroll
    for (int c = 0; c < 8; ++c) {
        const int i0 = __shfl(nbr, 4 * c + 2 * h,     32);
        const int i1 = __shfl(nbr, 4 * c + 2 * h + 1, 32);
        const float* r0 = feat + (long long)i0 * D_FEAT + m;
        const float* r1 = feat + (long long)i1 * D_FEAT + m;
#pragma unroll
        for (int d = 0; d < 8; ++d) {
            v2f a;                        // A: VGPR0 = row 2h, VGPR1 = row 2h+1 at dim 16d+m
            a[0] = r0[d * 16];            // 64B coalesced per half-wave
            a[1] = r1[d * 16];
            acc[d] = __builtin_amdgcn_wmma_f32_16x16x4_f32(
                false, a, false, bAll, (short)0, acc[d], false, false);
        }
    }

    // chunk 8: self row in K=0 slot only; weights {w, 0, 0, 0}
    {
        v2f bSelf;
        bSelf[0] = (lane < 16) ? w : 0.0f;   // lanes<16 hold K=0 (w), lanes>=16 hold K=2 (0)
        bSelf[1] = 0.0f;                     // K=1 / K=3 weights are 0
        const float* rs = feat + (long long)self * D_FEAT + m;
#pragma unroll
        for (int d = 0; d < 8; ++d) {
            v2f a;
            a[0] = (h == 0) ? rs[d * 16] : 0.0f;  // only K=0 slot carries data
            a[1] = 0.0f;
            acc[d] = __builtin_amdgcn_wmma_f32_16x16x4_f32(
                false, a, false, bSelf, (short)0, acc[d], false, false);
        }
    }

    // D layout: VGPR j, lanes 0-15 -> M=j ; lanes 16-31 -> M=8+j (all N columns equal).
    // Lane 0 stores dims 16d+0..7, lane 16 stores dims 16d+8..15, as 2x b128 each.
    if (m == 0) {
        float* op = out + (long long)b * D_FEAT + 8 * h;
#pragma unroll
        for (int d = 0; d < 8; ++d) {
            float4 lo = make_float4(acc[d][0], acc[d][1], acc[d][2], acc[d][3]);
            float4 hi = make_float4(acc[d][4], acc[d][5], acc[d][6], acc[d][7]);
            *(float4*)(op + d * 16)     = lo;
            *(float4*)(op + d * 16 + 4) = hi;
        }
    }
}

extern "C" void kernel_launch(void* const* d_in, const int* in_sizes, int n_in,
                              void* d_out, int out_size, void* d_ws, size_t ws_size,
                              hipStream_t stream) {
    const int*   nodes     = (const int*)d_in[0];
    const int*   neighbors = (const int*)d_in[1];
    // d_in[2] = num_sample (device scalar, fixed K=32 per reference shapes)
    const float* feat      = (const float*)d_in[3];
    float*       out       = (float*)d_out;

    const int batch = in_sizes[0];               // 20000
    const int wavesPerBlock = 8;                 // 256 threads / wave32
    const int blocks = (batch + wavesPerBlock - 1) / wavesPerBlock;

    MeanAggregator_45913200394470_kernel<<<blocks, 256, 0, stream>>>(
        nodes, neighbors, feat, out, batch);
}